// CeptaSSMLowRank_18674517803664
// MI455X (gfx1250) — compile-verified
//
#include <hip/hip_runtime.h>
#include <math.h>

// ---------------------------------------------------------------------------
// CeptaSSMLowRank for MI455X (gfx1250, wave32, WMMA + async LDS DMA).
// GEMMs via V_WMMA_F32_16X16X4_F32 (fp32 streamed from HBM -> memory bound,
// keep full precision). Global->LDS staging via GLOBAL_LOAD_ASYNC_TO_LDS_B128
// (ASYNCcnt), double-buffered against the WMMA K-loop.
// ---------------------------------------------------------------------------

typedef __attribute__((ext_vector_type(2))) float v2f;
typedef __attribute__((ext_vector_type(8))) float v8f;

#define P_DIM   1024
#define PR      128
#define B_DIM   8
#define T_DIM   4096
#define M_TOT   (B_DIM * T_DIM)   // 32768 rows
#define KT      128               // K chunk staged in LDS
#define LDK     (KT + 4)          // padded row (528 B, 16B aligned)
#define SCALE_U 0.08838834764831845f  // 1/sqrt(128)

__device__ __forceinline__ v8f wmma_f32(v2f a, v2f b, v8f c) {
    // 8 args: (neg_a, A, neg_b, B, c_mod, C, reuse_a, reuse_b)
    return __builtin_amdgcn_wmma_f32_16x16x4_f32(false, a, false, b,
                                                 (short)0, c, false, false);
}

// Async DMA: 16 bytes global -> LDS. lds_off = LDS byte address (low 32 bits
// of the generic shared pointer), goff = byte offset from 64-bit SGPR base.
__device__ __forceinline__ void async_g2l_b128(unsigned lds_off, unsigned goff,
                                               unsigned long long gbase) {
    asm volatile("global_load_async_to_lds_b128 %0, %1, %2"
                 :: "v"(lds_off), "v"(goff), "s"(gbase) : "memory");
}
__device__ __forceinline__ void wait_async_4() {
    asm volatile("s_wait_asynccnt 0x4" ::: "memory");
}
__device__ __forceinline__ void wait_async_0() {
    asm volatile("s_wait_asynccnt 0x0" ::: "memory");
}

// ---------------------------------------------------------------------------
// Kernel 1: per 16-row strip:
//   r = t @ V_r                      (kept in LDS)
//   u = (F*t) @ V_b * scale          (-> workspace)
//   a = clip(sigmoid(r @ W_l + b_l)) (-> workspace)
// Block: 256 threads = 8 waves; wave w owns output columns [16w, 16w+16).
// t and F K-chunks double-buffered in LDS via async DMA.
// ---------------------------------------------------------------------------
__global__ __launch_bounds__(256)
void ssm_proj_gate(const float* __restrict__ tin, const float* __restrict__ Fin,
                   const float* __restrict__ Vr,  const float* __restrict__ Vb,
                   const float* __restrict__ Wl,  const float* __restrict__ bl,
                   float* __restrict__ Ag, float* __restrict__ Ug)
{
    __shared__ __align__(16) float t_s[2][16][LDK];
    __shared__ __align__(16) float f_s[2][16][LDK];
    __shared__ __align__(16) float r_s[16][PR + 4];

    const int tid    = threadIdx.x;
    const int wave   = tid >> 5;
    const int lane   = tid & 31;
    const int lane_m = lane & 15;          // M for A-frag / N for B-frag
    const int lane_k = (lane >> 4) << 1;   // K sub-offset {0,2}
    const int hi     = lane >> 4;
    const int m0     = blockIdx.x * 16;
    const int n0     = wave * 16;

    const unsigned long long tb = (unsigned long long)(uintptr_t)tin;
    const unsigned long long fb = (unsigned long long)(uintptr_t)Fin;

    // Issue async stage of chunk into buffer `buf`.
    // 16 rows * 32 chunks(16B) = 512 per array; 2 per thread per array.
    auto stage = [&](int buf, int k0) {
        #pragma unroll
        for (int ii = 0; ii < 2; ++ii) {
            const int i = tid + ii * 256;
            const int r = i >> 5;              // 0..15
            const int c = (i & 31) << 2;       // float column, mult of 4
            const unsigned goff =
                (unsigned)(((m0 + r) * P_DIM + k0 + c) * 4);
            async_g2l_b128((unsigned)(uintptr_t)&t_s[buf][r][c], goff, tb);
            async_g2l_b128((unsigned)(uintptr_t)&f_s[buf][r][c], goff, fb);
        }
    };

    v8f cr = {0.f,0.f,0.f,0.f,0.f,0.f,0.f,0.f};
    v8f cu = {0.f,0.f,0.f,0.f,0.f,0.f,0.f,0.f};

    stage(0, 0);                         // prologue: fill buffer 0
    for (int k0 = 0; k0 < P_DIM; k0 += KT) {
        const int cur = (k0 / KT) & 1;
        if (k0 + KT < P_DIM) {
            stage(cur ^ 1, k0 + KT);     // prefetch next chunk
            wait_async_4();              // current chunk done; prefetch flies
        } else {
            wait_async_0();
        }
        __syncthreads();

        #pragma unroll 8
        for (int k = 0; k < KT; k += 4) {
            const int kk = k + lane_k;
            v2f av; av.x = t_s[cur][lane_m][kk];
                    av.y = t_s[cur][lane_m][kk + 1];
            v2f af; af.x = f_s[cur][lane_m][kk]     * av.x;
                    af.y = f_s[cur][lane_m][kk + 1] * av.y;
            const int kg = k0 + kk;
            v2f br; br.x = Vr[(size_t)kg * PR + n0 + lane_m];
                    br.y = Vr[(size_t)(kg + 1) * PR + n0 + lane_m];
            v2f bb; bb.x = Vb[(size_t)kg * PR + n0 + lane_m];
                    bb.y = Vb[(size_t)(kg + 1) * PR + n0 + lane_m];
            cr = wmma_f32(av, br, cr);
            cu = wmma_f32(af, bb, cu);
        }
        __syncthreads();                 // all waves done reading `cur`
    }

    // Spill r tile to LDS (C/D layout: vgpr v, lanes0-15 -> row v, 16-31 -> v+8)
    #pragma unroll
    for (int v = 0; v < 8; ++v) {
        const int rl = v + 8 * hi;
        r_s[rl][n0 + lane_m] = cr[v];
        Ug[(size_t)(m0 + rl) * PR + n0 + lane_m] = cu[v] * SCALE_U;
    }
    __syncthreads();

    // a = clip(sigmoid(r @ W_lambda + b_lambda)), K = 128
    v8f ca = {0.f,0.f,0.f,0.f,0.f,0.f,0.f,0.f};
    #pragma unroll 8
    for (int k = 0; k < PR; k += 4) {
        const int kk = k + lane_k;
        v2f av; av.x = r_s[lane_m][kk]; av.y = r_s[lane_m][kk + 1];
        v2f bw; bw.x = Wl[(size_t)kk * PR + n0 + lane_m];
                bw.y = Wl[(size_t)(kk + 1) * PR + n0 + lane_m];
        ca = wmma_f32(av, bw, ca);
    }
    const float bias = bl[n0 + lane_m];
    #pragma unroll
    for (int v = 0; v < 8; ++v) {
        const float x = ca[v] + bias;
        float g = 1.0f / (1.0f + __expf(-x));
        g = fminf(fmaxf(g, 0.01f), 0.995f);
        Ag[(size_t)(m0 + v + 8 * hi) * PR + n0 + lane_m] = g;
    }
}

// ---------------------------------------------------------------------------
// Kernel 2: sequential RMS-normalized scan — ONE WAVE per batch.
// Each lane owns 4 contiguous state components; reduction is pure in-wave
// shfl_xor (no barriers, no LDS). Loads are one coalesced b128 per lane.
// ---------------------------------------------------------------------------
__global__ __launch_bounds__(32)
void ssm_scan(const float* __restrict__ Ag, const float* __restrict__ Ug,
              float* __restrict__ S, float* __restrict__ cache_out)
{
    const int b    = blockIdx.x;
    const int lane = threadIdx.x;
    const size_t base = (size_t)b * T_DIM * PR + lane * 4;

    float4 s = make_float4(0.f, 0.f, 0.f, 0.f);

    for (int i = 0; i < T_DIM; ++i) {
        const size_t off = base + (size_t)i * PR;
        const float4 a = *(const float4*)(Ag + off);
        const float4 u = *(const float4*)(Ug + off);
        s.x = fmaf(a.x, s.x, u.x);
        s.y = fmaf(a.y, s.y, u.y);
        s.z = fmaf(a.z, s.z, u.z);
        s.w = fmaf(a.w, s.w, u.w);
        float x = s.x * s.x + s.y * s.y + s.z * s.z + s.w * s.w;
        #pragma unroll
        for (int o = 16; o > 0; o >>= 1) x += __shfl_xor(x, o, 32);
        const float inv = rsqrtf(x * (1.0f / (float)PR) + 1e-6f);
        s.x *= inv; s.y *= inv; s.z *= inv; s.w *= inv;
        *(float4*)(S + off) = s;
    }
    *(float4*)(cache_out + b * PR + lane * 4) = s;
}

// ---------------------------------------------------------------------------
// Kernel 3: t_tilde = S @ V_o + t   (K = 128)
// Grid: (M/16, P/128); block 256 threads = 8 waves, wave w -> 16-col subtile.
// S strip staged in LDS via async DMA.
// ---------------------------------------------------------------------------
__global__ __launch_bounds__(256)
void ssm_out_proj(const float* __restrict__ S, const float* __restrict__ Vo,
                  const float* __restrict__ tin, float* __restrict__ out)
{
    __shared__ __align__(16) float s_s[16][PR + 4];

    const int tid    = threadIdx.x;
    const int wave   = tid >> 5;
    const int lane   = tid & 31;
    const int lane_m = lane & 15;
    const int lane_k = (lane >> 4) << 1;
    const int hi     = lane >> 4;
    const int m0     = blockIdx.x * 16;
    const int n0     = blockIdx.y * 128 + wave * 16;

    // 16 rows * 8 chunks(16B) = 128 transfers; threads 0..127 issue one each.
    if (tid < 128) {
        const int r = tid >> 3;
        const int c = (tid & 7) << 2;
        const unsigned goff = (unsigned)(((m0 + r) * PR + c) * 4);
        async_g2l_b128((unsigned)(uintptr_t)&s_s[r][c], goff,
                       (unsigned long long)(uintptr_t)S);
    }
    wait_async_0();
    __syncthreads();

    v8f c = {0.f,0.f,0.f,0.f,0.f,0.f,0.f,0.f};
    #pragma unroll 8
    for (int k = 0; k < PR; k += 4) {
        const int kk = k + lane_k;
        v2f av; av.x = s_s[lane_m][kk]; av.y = s_s[lane_m][kk + 1];
        v2f bv; bv.x = Vo[(size_t)kk * P_DIM + n0 + lane_m];
                bv.y = Vo[(size_t)(kk + 1) * P_DIM + n0 + lane_m];
        c = wmma_f32(av, bv, c);
    }
    #pragma unroll
    for (int v = 0; v < 8; ++v) {
        const int rl = v + 8 * hi;
        const size_t idx = (size_t)(m0 + rl) * P_DIM + n0 + lane_m;
        out[idx] = c[v] + tin[idx];
    }
}

// ---------------------------------------------------------------------------
extern "C" void kernel_launch(void* const* d_in, const int* in_sizes, int n_in,
                              void* d_out, int out_size, void* d_ws, size_t ws_size,
                              hipStream_t stream) {
    const float* t  = (const float*)d_in[0];   // (B,T,P)
    const float* F  = (const float*)d_in[1];   // (B,T,P)
    const float* Vr = (const float*)d_in[2];   // (P,PR)
    const float* Vb = (const float*)d_in[3];   // (P,PR)
    const float* Vo = (const float*)d_in[4];   // (PR,P)
    const float* Wl = (const float*)d_in[5];   // (PR,PR)
    const float* bl = (const float*)d_in[6];   // (PR,)

    float* out   = (float*)d_out;                          // t_tilde (B,T,P)
    float* cache = out + (size_t)M_TOT * P_DIM;            // new_cache (B,PR)

    float* Ag = (float*)d_ws;                              // (M, PR) gates
    float* Ug = Ag + (size_t)M_TOT * PR;                   // (M, PR) inputs
    float* S  = Ug + (size_t)M_TOT * PR;                   // (M, PR) states

    ssm_proj_gate<<<dim3(M_TOT / 16), 256, 0, stream>>>(t, F, Vr, Vb, Wl, bl, Ag, Ug);
    ssm_scan<<<dim3(B_DIM), 32, 0, stream>>>(Ag, Ug, S, cache);
    ssm_out_proj<<<dim3(M_TOT / 16, P_DIM / 128), 256, 0, stream>>>(S, Vo, t, out);
}